// Net_18279380812273
// MI455X (gfx1250) — compile-verified
//
#include <hip/hip_runtime.h>
#include <hip/hip_bf16.h>
#include <math.h>

typedef __attribute__((ext_vector_type(2))) float v2f;
typedef __attribute__((ext_vector_type(8))) float v8f;

#define KTAB 36   // spline kernel table size (3*4*3)

// ---------------------------------------------------------------------------
// Spline basis + degree computation. One thread per edge.
// ---------------------------------------------------------------------------
__global__ __launch_bounds__(256)
void basis_deg_kernel(const float* __restrict__ pseudo,
                      const int* __restrict__ ei,
                      float* __restrict__ Bout,
                      int* __restrict__ WIout,
                      float* __restrict__ deg,
                      int E) {
  int e = blockIdx.x * blockDim.x + threadIdx.x;
  if (e >= E) return;

  const int ks[3]      = {3, 4, 3};
  const int open_[3]   = {1, 0, 1};
  const int strides[3] = {12, 3, 1};

  float w0[3], w1[3];
  int   i0[3], i1[3];
  #pragma unroll
  for (int d = 0; d < 3; ++d) {
    float u = pseudo[e * 3 + d];
    int   k = ks[d];
    float f;
    int a, b;
    if (open_[d]) {
      float v  = u * (float)(k - 1);
      float fl = floorf(v);
      f = v - fl;
      int ii = (int)fl;
      a = min(max(ii, 0), k - 1);
      b = min(a + 1, k - 1);
    } else {
      float v  = u * (float)k;
      float fl = floorf(v);
      f = v - fl;
      int ii = (int)fl;
      a = ii % k;       if (a < 0) a += k;
      b = (ii + 1) % k; if (b < 0) b += k;
    }
    w0[d] = 1.0f - f; w1[d] = f;
    i0[d] = a;        i1[d] = b;
  }

  #pragma unroll
  for (int s = 0; s < 8; ++s) {
    int b0 = s & 1, b1 = (s >> 1) & 1, b2 = (s >> 2) & 1;
    float w = (b0 ? w1[0] : w0[0]) * (b1 ? w1[1] : w0[1]) * (b2 ? w1[2] : w0[2]);
    int  wi = (b0 ? i1[0] : i0[0]) * strides[0]
            + (b1 ? i1[1] : i0[1]) * strides[1]
            + (b2 ? i1[2] : i0[2]) * strides[2];
    Bout[e * 8 + s]  = w;
    WIout[e * 8 + s] = wi;
  }

  int dst = ei[E + e];
  atomicAdd(&deg[dst], 1.0f);
}

__global__ __launch_bounds__(256)
void deginv_kernel(const float* __restrict__ deg, float* __restrict__ deg_inv, int N) {
  int i = blockIdx.x * blockDim.x + threadIdx.x;
  if (i < N) deg_inv[i] = 1.0f / fmaxf(deg[i], 1.0f);
}

// ---------------------------------------------------------------------------
// Scatter: A[(dst*36 + WI[e,s]) * Fin + f] += B[e,s] * x[src*Fin + f]
// One thread per (edge, feature).
// ---------------------------------------------------------------------------
__global__ __launch_bounds__(256)
void scatter_kernel(const float* __restrict__ x,
                    const float* __restrict__ Bb,
                    const int* __restrict__ WI,
                    const int* __restrict__ ei,
                    float* __restrict__ A,
                    int E, int Fin) {
  int idx = blockIdx.x * blockDim.x + threadIdx.x;
  if (idx >= E * Fin) return;
  int e = idx / Fin;
  int f = idx - e * Fin;
  int src = ei[e];
  int dst = ei[E + e];
  float xv = x[src * Fin + f];
  int base = dst * KTAB;
  #pragma unroll
  for (int s = 0; s < 8; ++s) {
    float w  = Bb[e * 8 + s];
    int   wi = WI[e * 8 + s];
    atomicAdd(&A[(base + wi) * Fin + f], w * xv);
  }
}

// A viewed as [N, rowlen]; scale each node-row by deg_inv[n].
__global__ __launch_bounds__(256)
void scale_rows_kernel(float* __restrict__ A, const float* __restrict__ deg_inv,
                       int N, int rowlen) {
  int i = blockIdx.x * blockDim.x + threadIdx.x;
  if (i >= N * rowlen) return;
  A[i] *= deg_inv[i / rowlen];
}

// ---------------------------------------------------------------------------
// Dual fp32 WMMA GEMM, LDS-staged.
//   C[M,Ncols] = A1[M,K1]@B1[K1,Ncols] + A2[M,K2]@B2[K2,Ncols] + bias (+ReLU)
// Row-major; ldb == Ncols. 256 threads = 8 waves: 2 M-groups x 4 N-groups.
// Block tile 64x128, K-step 8. Wave computes a 32x32 C tile (4 accumulators,
// 2 A frags x 2 B frags -> 8 wmma per staged tile). Main K loop is fully
// unmasked with pointer stepping; the single ragged K tile is a masked tail.
// Rows/cols are clamped for loads; only stores are guarded (EXEC all-ones).
// ---------------------------------------------------------------------------
#define MB 64
#define NB 128
#define KB 8

__global__ __launch_bounds__(256)
void gemm_dual_wmma(const float* __restrict__ A1, int lda1,
                    const float* __restrict__ B1, int K1,
                    const float* __restrict__ A2, int lda2,
                    const float* __restrict__ B2, int K2,
                    const float* __restrict__ bias,
                    float* __restrict__ C,
                    int M, int Ncols, int relu) {
  __shared__ float As[MB][KB + 2];   // padded stride: spread LDS banks
  __shared__ v2f   Bp[KB / 2][NB];   // B K-pairs: lane fragment = one b64 read

  const int tid  = threadIdx.x;
  const int wave = tid >> 5;
  const int lane = tid & 31;
  const int wm = wave & 1;      // M sub-group (0..1)
  const int wn = wave >> 1;     // N sub-group (0..3)
  const int g  = lane >> 4;     // lane half
  const int lm = lane & 15;

  const int tileM = blockIdx.y * MB;
  const int tileN = blockIdx.x * NB;

  // A staging: thread -> (row 0..63, k-pair 0/2/4/6)
  const int arl = tid >> 2;
  const int akk = (tid & 3) * 2;
  const int arowc = min(tileM + arl, M - 1);
  // B staging: thread -> (k-pair 0..3, cols bc and bc+64)
  const int bp  = tid >> 6;
  const int bc  = tid & 63;
  const int bcol0c = min(tileN + bc,      Ncols - 1);
  const int bcol1c = min(tileN + bc + 64, Ncols - 1);

  // compute-side fragment indices
  const int rl0 = wm * 32 + lm;        // local A row, acc row-tile 0
  const int rl1 = rl0 + 16;            // local A row, acc row-tile 1
  const int cl0 = wn * 32 + lm;        // local col, acc col-tile 0
  const int cl1 = cl0 + 16;            // local col, acc col-tile 1

  v8f acc00 = {0.f,0.f,0.f,0.f,0.f,0.f,0.f,0.f};
  v8f acc01 = acc00, acc10 = acc00, acc11 = acc00;

  auto compute_tile = [&]() {
    #pragma unroll
    for (int kk = 0; kk < KB; kk += 4) {
      int k0 = kk + 2 * g;                 // this lane-half's K pair base
      int p  = (kk >> 1) + g;              // B K-pair row
      v2f a0 = *(const v2f*)&As[rl0][k0];
      v2f a1 = *(const v2f*)&As[rl1][k0];
      v2f b0 = Bp[p][cl0];
      v2f b1 = Bp[p][cl1];
      acc00 = __builtin_amdgcn_wmma_f32_16x16x4_f32(false, a0, false, b0,
                                                    (short)0, acc00, false, false);
      acc10 = __builtin_amdgcn_wmma_f32_16x16x4_f32(false, a1, false, b0,
                                                    (short)0, acc10, false, false);
      acc01 = __builtin_amdgcn_wmma_f32_16x16x4_f32(false, a0, false, b1,
                                                    (short)0, acc01, false, false);
      acc11 = __builtin_amdgcn_wmma_f32_16x16x4_f32(false, a1, false, b1,
                                                    (short)0, acc11, false, false);
    }
  };

  const float* Aa = A1; const float* Bm = B1; int Kd = K1; int lda = lda1;
  for (int pass = 0; pass < 2; ++pass) {
    if (pass == 1) { Aa = A2; Bm = B2; Kd = K2; lda = lda2; }
    const int kfull = Kd & ~(KB - 1);      // full unmasked K tiles

    // pointer-stepping bases (row/col already clamped; K unmasked in main loop)
    const float* aptr  = Aa + (size_t)arowc * lda + akk;
    const float* bptrA = Bm + (size_t)(2 * bp) * Ncols + bcol0c;
    const float* bptrB = Bm + (size_t)(2 * bp) * Ncols + bcol1c;
    const size_t bstep = (size_t)KB * Ncols;

    for (int k = 0; k < kfull; k += KB) {
      float av0 = aptr[0];
      float av1 = aptr[1];
      float b00 = bptrA[0];
      float b01 = bptrA[Ncols];
      float b10 = bptrB[0];
      float b11 = bptrB[Ncols];
      __syncthreads();
      As[arl][akk]     = av0;
      As[arl][akk + 1] = av1;
      v2f bv0; bv0.x = b00; bv0.y = b01;
      v2f bv1; bv1.x = b10; bv1.y = b11;
      Bp[bp][bc]      = bv0;
      Bp[bp][bc + 64] = bv1;
      __syncthreads();
      compute_tile();
      aptr  += KB;
      bptrA += bstep;
      bptrB += bstep;
    }

    if (kfull < Kd) {                      // single masked tail tile
      int ka0 = kfull + akk, ka1 = ka0 + 1;
      float av0 = Aa[(size_t)arowc * lda + min(ka0, Kd - 1)];
      float av1 = Aa[(size_t)arowc * lda + min(ka1, Kd - 1)];
      int kb0 = kfull + 2 * bp, kb1 = kb0 + 1;
      const float* bm0 = Bm + (size_t)min(kb0, Kd - 1) * Ncols;
      const float* bm1 = Bm + (size_t)min(kb1, Kd - 1) * Ncols;
      float b00 = bm0[bcol0c], b01 = bm1[bcol0c];
      float b10 = bm0[bcol1c], b11 = bm1[bcol1c];
      __syncthreads();
      As[arl][akk]     = (ka0 < Kd) ? av0 : 0.f;
      As[arl][akk + 1] = (ka1 < Kd) ? av1 : 0.f;
      v2f bv0; bv0.x = (kb0 < Kd) ? b00 : 0.f; bv0.y = (kb1 < Kd) ? b01 : 0.f;
      v2f bv1; bv1.x = (kb0 < Kd) ? b10 : 0.f; bv1.y = (kb1 < Kd) ? b11 : 0.f;
      Bp[bp][bc]      = bv0;
      Bp[bp][bc + 64] = bv1;
      __syncthreads();
      compute_tile();
    }
  }

  // ---- epilogue ----
  const int col0 = tileN + cl0;
  const int col1 = tileN + cl1;
  float bv0 = bias ? bias[min(col0, Ncols - 1)] : 0.f;
  float bv1 = bias ? bias[min(col1, Ncols - 1)] : 0.f;
  bool c0ok = col0 < Ncols;
  bool c1ok = col1 < Ncols;
  #pragma unroll
  for (int v = 0; v < 8; ++v) {
    int mr = v + 8 * g;                       // C row within 16-row tile
    int r0 = tileM + wm * 32 + mr;
    int r1 = r0 + 16;
    float o00 = acc00[v] + bv0, o01 = acc01[v] + bv1;
    float o10 = acc10[v] + bv0, o11 = acc11[v] + bv1;
    if (relu) {
      o00 = fmaxf(o00, 0.f); o01 = fmaxf(o01, 0.f);
      o10 = fmaxf(o10, 0.f); o11 = fmaxf(o11, 0.f);
    }
    if (r0 < M) {
      if (c0ok) C[(size_t)r0 * Ncols + col0] = o00;
      if (c1ok) C[(size_t)r0 * Ncols + col1] = o01;
    }
    if (r1 < M) {
      if (c0ok) C[(size_t)r1 * Ncols + col0] = o10;
      if (c1ok) C[(size_t)r1 * Ncols + col1] = o11;
    }
  }
}

// ---------------------------------------------------------------------------
// In-place row log-softmax. One block per row.
// ---------------------------------------------------------------------------
__global__ __launch_bounds__(256)
void log_softmax_kernel(float* __restrict__ out, int Ncols) {
  __shared__ float red[256];
  float* p = out + (size_t)blockIdx.x * Ncols;

  float mx = -INFINITY;
  for (int c = threadIdx.x; c < Ncols; c += blockDim.x) mx = fmaxf(mx, p[c]);
  red[threadIdx.x] = mx;
  __syncthreads();
  for (int s = 128; s > 0; s >>= 1) {
    if (threadIdx.x < s) red[threadIdx.x] = fmaxf(red[threadIdx.x], red[threadIdx.x + s]);
    __syncthreads();
  }
  mx = red[0];
  __syncthreads();

  float sum = 0.f;
  for (int c = threadIdx.x; c < Ncols; c += blockDim.x) sum += expf(p[c] - mx);
  red[threadIdx.x] = sum;
  __syncthreads();
  for (int s = 128; s > 0; s >>= 1) {
    if (threadIdx.x < s) red[threadIdx.x] += red[threadIdx.x + s];
    __syncthreads();
  }
  float lse = logf(red[0]) + mx;

  for (int c = threadIdx.x; c < Ncols; c += blockDim.x) p[c] = p[c] - lse;
}

// ---------------------------------------------------------------------------
// Host launcher
// ---------------------------------------------------------------------------
static inline int ceil_div(int a, int b) { return (a + b - 1) / b; }

extern "C" void kernel_launch(void* const* d_in, const int* in_sizes, int n_in,
                              void* d_out, int out_size, void* d_ws, size_t ws_size,
                              hipStream_t stream) {
  const float* x     = (const float*)d_in[0];
  const float* ps    = (const float*)d_in[1];
  const int*   ei    = (const int*)  d_in[2];
  const float* W1    = (const float*)d_in[3];
  const float* root1 = (const float*)d_in[4];
  const float* b1    = (const float*)d_in[5];
  const float* W2    = (const float*)d_in[6];
  const float* root2 = (const float*)d_in[7];
  const float* b2    = (const float*)d_in[8];
  const float* W3    = (const float*)d_in[9];
  const float* root3 = (const float*)d_in[10];
  const float* b3    = (const float*)d_in[11];
  const float* lw1   = (const float*)d_in[12];
  const float* lb1   = (const float*)d_in[13];
  const float* lw2   = (const float*)d_in[14];
  const float* lb2   = (const float*)d_in[15];

  const int N = in_sizes[0];        // x: [N,1]
  const int E = in_sizes[1] / 3;    // pseudo: [E,3]
  const int NC = 6890;

  // Workspace layout
  char* w = (char*)d_ws;
  float* Bbuf  = (float*)w; w += (size_t)E * 8 * sizeof(float);
  int*   WIbuf = (int*)w;   w += (size_t)E * 8 * sizeof(int);
  float* deg   = (float*)w; w += (size_t)N * sizeof(float);
  float* dinv  = (float*)w; w += (size_t)N * sizeof(float);
  float* Abuf  = (float*)w; w += (size_t)N * KTAB * 64 * sizeof(float);  // max Fin=64
  float* h1    = (float*)w; w += (size_t)N * 32 * sizeof(float);
  float* h2    = (float*)w; w += (size_t)N * 64 * sizeof(float);
  float* h3    = (float*)w; w += (size_t)N * 128 * sizeof(float);
  float* h4    = (float*)w; w += (size_t)N * 256 * sizeof(float);

  float* logits = (float*)d_out;

  // 1) basis + degree
  hipMemsetAsync(deg, 0, (size_t)N * sizeof(float), stream);
  basis_deg_kernel<<<ceil_div(E, 256), 256, 0, stream>>>(ps, ei, Bbuf, WIbuf, deg, E);
  deginv_kernel<<<ceil_div(N, 256), 256, 0, stream>>>(deg, dinv, N);

  // Spline conv layers: {input, Fin, W, root, bias, Fout, output}
  struct Layer { const float* in; int Fin; const float* W; const float* root;
                 const float* bias; int Fout; float* out; };
  const Layer layers[3] = {
    { x,  1,  W1, root1, b1, 32,  h1 },
    { h1, 32, W2, root2, b2, 64,  h2 },
    { h2, 64, W3, root3, b3, 128, h3 },
  };

  for (int l = 0; l < 3; ++l) {
    const Layer& L = layers[l];
    const int rowlen = KTAB * L.Fin;                 // K1 of the GEMM
    hipMemsetAsync(Abuf, 0, (size_t)N * rowlen * sizeof(float), stream);
    scatter_kernel<<<ceil_div(E * L.Fin, 256), 256, 0, stream>>>(
        L.in, Bbuf, WIbuf, ei, Abuf, E, L.Fin);
    scale_rows_kernel<<<ceil_div(N * rowlen, 256), 256, 0, stream>>>(
        Abuf, dinv, N, rowlen);
    dim3 grid(ceil_div(L.Fout, NB), ceil_div(N, MB));
    gemm_dual_wmma<<<grid, 256, 0, stream>>>(
        Abuf, rowlen, L.W, rowlen,      // A1 @ W  (deg_inv already folded in)
        L.in, L.Fin, L.root, L.Fin,     // x  @ root
        L.bias, L.out, N, L.Fout, /*relu=*/1);
  }

  // 4) h4 = relu(h3 @ lw1 + lb1)
  {
    dim3 grid(ceil_div(256, NB), ceil_div(N, MB));
    gemm_dual_wmma<<<grid, 256, 0, stream>>>(
        h3, 128, lw1, 128, h3, 128, lw1, 0, lb1, h4, N, 256, 1);
  }

  // 5) logits = h4 @ lw2 + lb2
  {
    dim3 grid(ceil_div(NC, NB), ceil_div(N, MB));
    gemm_dual_wmma<<<grid, 256, 0, stream>>>(
        h4, 256, lw2, 256, h4, 256, lw2, 0, lb2, logits, N, NC, 0);
  }

  // 6) log-softmax in place
  log_softmax_kernel<<<N, 256, 0, stream>>>(logits, NC);
}